// ShinigamiSSMNeuron_4105988735913
// MI455X (gfx1250) — compile-verified
//
#include <hip/hip_runtime.h>
#include <math.h>

#define BB 2
#define TT 512
#define DD 256
#define KW 8
#define BT (BB * TT)      // 1024
#define D2 (2 * DD)       // 512

typedef __attribute__((ext_vector_type(2))) float v2f;
typedef __attribute__((ext_vector_type(8))) float v8f;

// ---------------------------------------------------------------------------
// Kernel 1: fused = x + 0.3*pg ; depthwise causal conv (left pad K-1) ;
//           combined = fused + conv + bias.   One block per (b,t), thread = d.
// ---------------------------------------------------------------------------
__global__ void k_fuse_conv(const float* __restrict__ x,
                            const float* __restrict__ pg,
                            const float* __restrict__ cw,   // [D,1,K]
                            const float* __restrict__ cb,   // [D]
                            float* __restrict__ combined) {
    int bt = blockIdx.x;           // b*T + t
    int t  = bt % TT;
    int d  = threadIdx.x;
    float acc  = 0.0f;
    float fu_t = 0.0f;
#pragma unroll
    for (int k = 0; k < KW; ++k) {
        int tt = t - (KW - 1) + k;
        float fu = 0.0f;
        if (tt >= 0) {
            size_t idx = (size_t)(bt - (KW - 1) + k) * DD + d;
            fu = x[idx] + 0.3f * pg[idx];
        }
        acc = fmaf(cw[d * KW + k], fu, acc);
        if (k == KW - 1) fu_t = fu;
    }
    combined[(size_t)bt * DD + d] = fu_t + acc + cb[d];
}

// ---------------------------------------------------------------------------
// Kernel 2: gates GEMM via V_WMMA_F32_16X16X4_F32.
//   f = sigmoid(combined @ Wf^T + bf); i = combined @ Wi^T + bi
//   u = (1-f)*i ; lf = log(f + 1e-6)
// One wave (32 lanes) per 16x16 output tile; A fragment shared by both GEMMs.
// ---------------------------------------------------------------------------
__global__ void k_gates(const float* __restrict__ comb,
                        const float* __restrict__ Wf, const float* __restrict__ bf,
                        const float* __restrict__ Wi, const float* __restrict__ bi,
                        float* __restrict__ u, float* __restrict__ lf) {
    int lane = threadIdx.x;            // 0..31, full wave active
    int row0 = blockIdx.x * 16;        // token rows
    int col0 = blockIdx.y * 16;        // output features
    int m  = lane & 15;
    int kh = (lane >> 4) << 1;         // 0 (lanes 0-15) or 2 (lanes 16-31)
    const float* arow = comb + (size_t)(row0 + m) * DD;   // A: 16 x 4 slab walker
    const float* frow = Wf   + (size_t)(col0 + m) * DD;   // B: Wf[col, k]
    const float* irow = Wi   + (size_t)(col0 + m) * DD;
    v8f cf = {};
    v8f ci = {};
    for (int k0 = 0; k0 < DD; k0 += 4) {
        v2f a;  a.x  = arow[k0 + kh]; a.y  = arow[k0 + kh + 1];
        v2f bv; bv.x = frow[k0 + kh]; bv.y = frow[k0 + kh + 1];
        v2f cv; cv.x = irow[k0 + kh]; cv.y = irow[k0 + kh + 1];
        cf = __builtin_amdgcn_wmma_f32_16x16x4_f32(false, a, false, bv,
                                                   (short)0, cf, false, false);
        ci = __builtin_amdgcn_wmma_f32_16x16x4_f32(false, a, false, cv,
                                                   (short)0, ci, false, false);
    }
    int n  = lane & 15;
    int mb = (lane >> 4) * 8;          // C/D layout: vgpr r -> M=r or M=r+8
#pragma unroll
    for (int r = 0; r < 8; ++r) {
        int row = row0 + mb + r;
        int col = col0 + n;
        float fv = cf[r] + bf[col];
        float iv = ci[r] + bi[col];
        float f  = 1.0f / (1.0f + expf(-fv));
        size_t idx = (size_t)row * DD + col;
        u[idx]  = (1.0f - f) * iv;
        lf[idx] = logf(f + 1e-6f);
    }
}

// ---------------------------------------------------------------------------
// Kernel 3: per-channel linear recurrence replacing the T x T decay einsum:
//   h[t] = exp(lf[t]) * h[t-1] + u[t]
// One thread per (b,d) channel; loads are coalesced across d each step.
// ---------------------------------------------------------------------------
__global__ void k_scan(const float* __restrict__ u,
                       const float* __restrict__ lf,
                       float* __restrict__ h) {
    int tid = blockIdx.x * blockDim.x + threadIdx.x;   // 0 .. B*D-1
    int b = tid / DD;
    int d = tid % DD;
    size_t base = (size_t)b * TT * DD + d;
    float hc = 0.0f;
    for (int t = 0; t < TT; ++t) {
        size_t idx = base + (size_t)t * DD;
        hc = fmaf(expf(lf[idx]), hc, u[idx]);
        h[idx] = hc;
    }
}

// ---------------------------------------------------------------------------
// Kernel 4: g = gelu(h @ W1^T + b1), exact gelu (erf). Out [BT, 2D].
// ---------------------------------------------------------------------------
__global__ void k_mlp1(const float* __restrict__ h,
                       const float* __restrict__ W1, const float* __restrict__ b1,
                       float* __restrict__ g) {
    int lane = threadIdx.x;
    int row0 = blockIdx.x * 16;
    int col0 = blockIdx.y * 16;
    int m  = lane & 15;
    int kh = (lane >> 4) << 1;
    const float* arow = h  + (size_t)(row0 + m) * DD;
    const float* brow = W1 + (size_t)(col0 + m) * DD;
    v8f c = {};
    for (int k0 = 0; k0 < DD; k0 += 4) {
        v2f a;  a.x  = arow[k0 + kh]; a.y  = arow[k0 + kh + 1];
        v2f bv; bv.x = brow[k0 + kh]; bv.y = brow[k0 + kh + 1];
        c = __builtin_amdgcn_wmma_f32_16x16x4_f32(false, a, false, bv,
                                                  (short)0, c, false, false);
    }
    int n  = lane & 15;
    int mb = (lane >> 4) * 8;
#pragma unroll
    for (int r = 0; r < 8; ++r) {
        int row = row0 + mb + r;
        int col = col0 + n;
        float v = c[r] + b1[col];
        g[(size_t)row * D2 + col] = 0.5f * v * (1.0f + erff(v * 0.70710678118654752f));
    }
}

// ---------------------------------------------------------------------------
// Kernel 5: y = h + g @ W2^T + b2. K = 512.
// ---------------------------------------------------------------------------
__global__ void k_mlp2(const float* __restrict__ g,
                       const float* __restrict__ W2, const float* __restrict__ b2,
                       const float* __restrict__ h,
                       float* __restrict__ y) {
    int lane = threadIdx.x;
    int row0 = blockIdx.x * 16;
    int col0 = blockIdx.y * 16;
    int m  = lane & 15;
    int kh = (lane >> 4) << 1;
    const float* arow = g  + (size_t)(row0 + m) * D2;
    const float* brow = W2 + (size_t)(col0 + m) * D2;
    v8f c = {};
    for (int k0 = 0; k0 < D2; k0 += 4) {
        v2f a;  a.x  = arow[k0 + kh]; a.y  = arow[k0 + kh + 1];
        v2f bv; bv.x = brow[k0 + kh]; bv.y = brow[k0 + kh + 1];
        c = __builtin_amdgcn_wmma_f32_16x16x4_f32(false, a, false, bv,
                                                  (short)0, c, false, false);
    }
    int n  = lane & 15;
    int mb = (lane >> 4) * 8;
#pragma unroll
    for (int r = 0; r < 8; ++r) {
        int row = row0 + mb + r;
        int col = col0 + n;
        size_t idx = (size_t)row * DD + col;
        y[idx] = h[idx] + c[r] + b2[col];
    }
}

// ---------------------------------------------------------------------------
// Kernel 6: LayerNorm over D (population variance), one row per block.
// ---------------------------------------------------------------------------
__global__ void k_ln(const float* __restrict__ y,
                     const float* __restrict__ gamma,
                     const float* __restrict__ beta,
                     float* __restrict__ out) {
    __shared__ float red[DD];
    int row = blockIdx.x;
    int d   = threadIdx.x;
    float v = y[(size_t)row * DD + d];
    red[d] = v;
    __syncthreads();
    for (int s = DD / 2; s > 0; s >>= 1) {
        if (d < s) red[d] += red[d + s];
        __syncthreads();
    }
    float mu = red[0] * (1.0f / DD);
    __syncthreads();
    float dv = v - mu;
    red[d] = dv * dv;
    __syncthreads();
    for (int s = DD / 2; s > 0; s >>= 1) {
        if (d < s) red[d] += red[d + s];
        __syncthreads();
    }
    float var = red[0] * (1.0f / DD);
    out[(size_t)row * DD + d] = dv * rsqrtf(var + 1e-5f) * gamma[d] + beta[d];
}

// ---------------------------------------------------------------------------
extern "C" void kernel_launch(void* const* d_in, const int* in_sizes, int n_in,
                              void* d_out, int out_size, void* d_ws, size_t ws_size,
                              hipStream_t stream) {
    const float* x    = (const float*)d_in[0];
    const float* pg   = (const float*)d_in[1];
    const float* cw   = (const float*)d_in[2];
    const float* cb   = (const float*)d_in[3];
    const float* Wf   = (const float*)d_in[4];
    const float* bf   = (const float*)d_in[5];
    const float* Wi   = (const float*)d_in[6];
    const float* bi   = (const float*)d_in[7];
    const float* W1   = (const float*)d_in[8];
    const float* b1   = (const float*)d_in[9];
    const float* W2   = (const float*)d_in[10];
    const float* b2   = (const float*)d_in[11];
    const float* gam  = (const float*)d_in[12];
    const float* bet  = (const float*)d_in[13];

    // Workspace layout (floats). BT*D = 256K floats, BT*2D = 512K floats.
    // combined: [0, 256K)   -- dead after k_gates
    // u       : [256K,512K) -- dead after k_scan
    // lf      : [512K,768K) -- dead after k_scan
    // h       : [768K,1M)   -- live through k_mlp2
    // g       : [0, 512K)   -- reuses combined+u after both are dead
    // y       : [512K,768K) -- reuses lf region
    float* ws       = (float*)d_ws;
    float* combined = ws;
    float* u        = ws + 256 * 1024;
    float* lf       = ws + 512 * 1024;
    float* h        = ws + 768 * 1024;
    float* g        = ws;                // 512K floats
    float* y        = ws + 512 * 1024;   // 256K floats
    float* out      = (float*)d_out;

    k_fuse_conv<<<dim3(BT), dim3(DD), 0, stream>>>(x, pg, cw, cb, combined);
    k_gates    <<<dim3(BT / 16, DD / 16), dim3(32), 0, stream>>>(combined, Wf, bf, Wi, bi, u, lf);
    k_scan     <<<dim3((BB * DD + 255) / 256), dim3(256), 0, stream>>>(u, lf, h);
    k_mlp1     <<<dim3(BT / 16, D2 / 16), dim3(32), 0, stream>>>(h, W1, b1, g);
    k_mlp2     <<<dim3(BT / 16, DD / 16), dim3(32), 0, stream>>>(g, W2, b2, h, y);
    k_ln       <<<dim3(BT), dim3(DD), 0, stream>>>(y, gam, bet, out);
}